// EquivariantAttention_60687887892713
// MI455X (gfx1250) — compile-verified
//
#include <hip/hip_runtime.h>

typedef __attribute__((ext_vector_type(2))) float v2f;
typedef __attribute__((ext_vector_type(4))) float v4f;
typedef __attribute__((ext_vector_type(8))) float v8f;
typedef __attribute__((ext_vector_type(4))) int   v4i;

// Problem constants (fixed by the reference)
#define NNODE   8192
#define NEDGE   65536
#define GKPAD   2112          // 2048 (hidden x tvec) + 32 (bias fold) + 32 zero pad
#define BSTRIDE 104           // LDS dword stride of one W2T row (bank-conflict-free)
#define HSTRIDE 68            // LDS dword stride of one hidden row (64 + 1.0 slot + 0.0 slot + pad)
#define TSTRIDE 34            // LDS dword stride of one tvec row
#define CK      64            // K-rows per staged chunk
#define NCHUNK  33            // 2112 / 64

// ---- CDNA5 async global->LDS staging (ASYNCcnt-tracked), with safe fallback ----
#if defined(__AMDGCN__) && __has_builtin(__builtin_amdgcn_global_load_async_to_lds_b128)
#define HAVE_ASYNC_LDS 1
// (global v4i* src, lds v4i* dst, imm offset, imm cpol) -> global_load_async_to_lds_b128
#define ASYNC_CP(gp, lp)                                                        \
    __builtin_amdgcn_global_load_async_to_lds_b128(                             \
        (__attribute__((address_space(1))) v4i*)(gp),                           \
        (__attribute__((address_space(3))) v4i*)(lp), 0, 0)
#if __has_builtin(__builtin_amdgcn_s_wait_asynccnt)
#define WAIT_ASYNC(n) __builtin_amdgcn_s_wait_asynccnt(n)
#else
#define WAIT_ASYNC(n) asm volatile("s_wait_asynccnt %0" ::"i"(n) : "memory")
#endif
#endif

__device__ __forceinline__ unsigned ordf(float f) {
    unsigned u = __float_as_uint(f);
    return (u >> 31) ? ~u : (u | 0x80000000u);
}
__device__ __forceinline__ float unordf(unsigned v) {
    unsigned u = (v & 0x80000000u) ? (v ^ 0x80000000u) : ~v;
    return __uint_as_float(u);
}

// ---------------------------------------------------------------------------
// K-major re-layout of rw_w2 (64 x 3072) -> W2T (2112 x 96), bias folded in.
// ---------------------------------------------------------------------------
__global__ __launch_bounds__(256) void k_prep(const float* __restrict__ w2,
                                              const float* __restrict__ bias2,
                                              float* __restrict__ w2t) {
    int idx = blockIdx.x * 256 + threadIdx.x;          // < 2112*96 = 202752
    int kk = idx / 96, i = idx - kk * 96;
    float v;
    if (kk < 2048)      { int h = kk >> 5, j = kk & 31; v = w2[h * 3072 + i * 32 + j]; }
    else if (kk < 2080) { int j = kk - 2048;            v = bias2[i * 32 + j]; }
    else                { v = 0.f; }
    w2t[idx] = v;
}

__global__ __launch_bounds__(256) void k_init(float* __restrict__ nodeSum,
                                              float* __restrict__ agg,
                                              unsigned* __restrict__ nodeMax) {
    int i = blockIdx.x * 256 + threadIdx.x;            // < N*64
    agg[i] = 0.f;
    if (i < NNODE * 4) { nodeSum[i] = 0.f; nodeMax[i] = 0x007FFFFFu; /* ord(-inf) */ }
}

// ---------------------------------------------------------------------------
// Main fused kernel: MLP + basis contractions + attention scores.
// 8 waves/block, one 16-edge WMMA M-tile per wave, 128 edges per block.
// ---------------------------------------------------------------------------
__global__ __launch_bounds__(256) void k_conv(
    const float* __restrict__ ef, const float* __restrict__ f,
    const float* __restrict__ basis1, const float* __restrict__ basis2,
    const int* __restrict__ srcIdx, const int* __restrict__ dstIdx,
    const float* __restrict__ w1, const float* __restrict__ b1,
    const float* __restrict__ w2t,
    float* __restrict__ vvOut, float* __restrict__ scOut,
    unsigned* __restrict__ nodeMax)
{
    extern __shared__ float smem[];
    float* shH = smem;                        // 8*16*68  = 8704 dwords
    float* shT = shH + 8 * 16 * HSTRIDE;      // 8*16*34  = 4352 dwords
    float* shB = shT + 8 * 16 * TSTRIDE;      // 2*64*104 = 13312 dwords (reused as t scratch)

    const int tid  = threadIdx.x;
    const int wave = tid >> 5;
    const int lane = tid & 31;
    const int grp  = lane >> 4;               // which half of the wave (A/B K-split)
    const int row  = lane & 15;               // edge row within the 16x tile / N column
    const int edge = blockIdx.x * 128 + wave * 16 + row;

    // ---- Phase 1: hidden = relu(ef @ W1 + b1) via V_WMMA_F32_16X16X4_F32 ----
    v2f aef[8];
    {
        const float* p = ef + (size_t)edge * 32 + grp * 2;
        #pragma unroll
        for (int ks = 0; ks < 8; ++ks) { aef[ks].x = p[ks * 4]; aef[ks].y = p[ks * 4 + 1]; }
    }
    #pragma unroll
    for (int nt = 0; nt < 4; ++nt) {
        float bias = b1[nt * 16 + row];
        v8f acc;
        #pragma unroll
        for (int v = 0; v < 8; ++v) acc[v] = bias;
        #pragma unroll
        for (int ks = 0; ks < 8; ++ks) {
            int kr = ks * 4 + grp * 2;
            v2f bb;
            bb.x = w1[kr * 64 + nt * 16 + row];
            bb.y = w1[(kr + 1) * 64 + nt * 16 + row];
            acc = __builtin_amdgcn_wmma_f32_16x16x4_f32(false, aef[ks], false, bb,
                                                        (short)0, acc, false, false);
        }
        #pragma unroll
        for (int v = 0; v < 8; ++v) {
            int M = v + grp * 8;
            float hv = acc[v] > 0.f ? acc[v] : 0.f;
            shH[(wave * 16 + M) * HSTRIDE + nt * 16 + row] = hv;
        }
    }
    if (grp == 0) {                            // bias-fold slot and zero-pad slot
        shH[(wave * 16 + row) * HSTRIDE + 64] = 1.0f;
        shH[(wave * 16 + row) * HSTRIDE + 65] = 0.0f;
    }

    // ---- Phase 2: tvec[e, m*2+k] = sum_d f[src,m,d] * basis1[e,d,k] ----
    {
        int s = srcIdx[edge];
        const float* bp = basis1 + (size_t)edge * 8;
        v4f bA = *(const v4f*)(bp);            // d0k0 d0k1 d1k0 d1k1
        v4f bB = *(const v4f*)(bp + 4);        // d2k0 d2k1 d3k0 d3k1
        #pragma unroll
        for (int mi = 0; mi < 8; ++mi) {
            int m = grp * 8 + mi;
            v4f fv = *(const v4f*)(f + (size_t)s * 64 + m * 4);
            v2f tw;
            tw.x = fv.x * bA.x + fv.y * bA.z + fv.z * bB.x + fv.w * bB.z;
            tw.y = fv.x * bA.y + fv.y * bA.w + fv.z * bB.y + fv.w * bB.w;
            *(v2f*)(shT + (wave * 16 + row) * TSTRIDE + m * 2) = tw;
        }
    }
    __syncthreads();

    // ---- Phase 3: t = g @ W2T, K = 2112, A built on the fly, B double-buffered ----
    v8f acc[6];
    #pragma unroll
    for (int nt = 0; nt < 6; ++nt)
        #pragma unroll
        for (int v = 0; v < 8; ++v) acc[nt][v] = 0.f;

    int buf = 0;
    const float* hR = shH + (wave * 16 + row) * HSTRIDE;
    const float* tR = shT + (wave * 16 + row) * TSTRIDE;

#ifdef HAVE_ASYNC_LDS
    // --- CDNA5 path: global_load_async_to_lds_b128 + s_wait_asynccnt ---
    {
        #pragma unroll
        for (int it = 0; it < 6; ++it) {        // stage chunk 0 (6 x b128 per thread)
            int g = tid + it * 256;
            ASYNC_CP(w2t + (g / 24) * 96 + (g % 24) * 4,
                     shB + (g / 24) * BSTRIDE + (g % 24) * 4);
        }
    }
    for (int ch = 0; ch < NCHUNK; ++ch) {
        if (ch + 1 < NCHUNK) {                  // issue next chunk into the other buffer
            const float* gs = w2t + (size_t)(ch + 1) * (CK * 96);
            float*       ld = shB + (buf ^ 1) * (CK * BSTRIDE);
            #pragma unroll
            for (int it = 0; it < 6; ++it) {
                int g = tid + it * 256;
                ASYNC_CP(gs + (g / 24) * 96 + (g % 24) * 4,
                         ld + (g / 24) * BSTRIDE + (g % 24) * 4);
            }
            if (ch + 2 < NCHUNK)                // -> global_prefetch_b8 for chunk after next
                __builtin_prefetch(w2t + (size_t)(ch + 2) * (CK * 96) + (tid << 4), 0, 1);
            WAIT_ASYNC(6);                      // async completes in order: chunk ch landed
        } else {
            WAIT_ASYNC(0);
        }
        __syncthreads();                        // every wave's chunk ch is in LDS
        const float* bB = shB + buf * (CK * BSTRIDE);
        const int kbase = ch * CK + grp * 2;
        #pragma unroll
        for (int ks = 0; ks < 16; ++ks) {
            int kk0  = kbase + ks * 4;
            float hv = hR[kk0 >> 5];
            v2f  tv  = *(const v2f*)(tR + (kk0 & 31));
            v2f  A;  A.x = hv * tv.x; A.y = hv * tv.y;
            int brow = ks * 4 + grp * 2;
            #pragma unroll
            for (int nt = 0; nt < 6; ++nt) {
                v2f B;
                B.x = bB[brow * BSTRIDE + nt * 16 + row];
                B.y = bB[(brow + 1) * BSTRIDE + nt * 16 + row];
                acc[nt] = __builtin_amdgcn_wmma_f32_16x16x4_f32(false, A, false, B,
                                                                (short)0, acc[nt], false, false);
            }
        }
        __syncthreads();    // all reads of 'buf' done before anyone async-writes it next iter
        buf ^= 1;
    }
#else
    // --- Fallback: register-staged software pipeline (proven path) ---
    {
        #pragma unroll
        for (int it = 0; it < 6; ++it) {
            int g = tid + it * 256;
            *(v4f*)(shB + (g / 24) * BSTRIDE + (g % 24) * 4) =
                *(const v4f*)(w2t + (g / 24) * 96 + (g % 24) * 4);
        }
    }
    for (int ch = 0; ch < NCHUNK; ++ch) {
        __syncthreads();
        v4f stg[6];
        if (ch + 1 < NCHUNK) {
            const float* gs = w2t + (size_t)(ch + 1) * (CK * 96);
            #pragma unroll
            for (int it = 0; it < 6; ++it) {
                int g = tid + it * 256;
                stg[it] = *(const v4f*)(gs + (g / 24) * 96 + (g % 24) * 4);
            }
            if (ch + 2 < NCHUNK)
                __builtin_prefetch(w2t + (size_t)(ch + 2) * (CK * 96) + (tid << 4), 0, 1);
        }
        const float* bB = shB + buf * (CK * BSTRIDE);
        const int kbase = ch * CK + grp * 2;
        #pragma unroll
        for (int ks = 0; ks < 16; ++ks) {
            int kk0  = kbase + ks * 4;
            float hv = hR[kk0 >> 5];
            v2f  tv  = *(const v2f*)(tR + (kk0 & 31));
            v2f  A;  A.x = hv * tv.x; A.y = hv * tv.y;
            int brow = ks * 4 + grp * 2;
            #pragma unroll
            for (int nt = 0; nt < 6; ++nt) {
                v2f B;
                B.x = bB[brow * BSTRIDE + nt * 16 + row];
                B.y = bB[(brow + 1) * BSTRIDE + nt * 16 + row];
                acc[nt] = __builtin_amdgcn_wmma_f32_16x16x4_f32(false, A, false, B,
                                                                (short)0, acc[nt], false, false);
            }
        }
        if (ch + 1 < NCHUNK) {
            float* ld = shB + (buf ^ 1) * (CK * BSTRIDE);
            #pragma unroll
            for (int it = 0; it < 6; ++it) {
                int g = tid + it * 256;
                *(v4f*)(ld + (g / 24) * BSTRIDE + (g % 24) * 4) = stg[it];
            }
        }
        buf ^= 1;
    }
#endif
    __syncthreads();                           // everyone done with shB -> reuse as t scratch

    // ---- Phase 4: park t (16x96 per wave), then conv epilogue + attention scores ----
    {
        float* tsc = shB + wave * 1600;        // 16 rows x 100-dword stride
        #pragma unroll
        for (int nt = 0; nt < 6; ++nt)
            #pragma unroll
            for (int v = 0; v < 8; ++v)
                tsc[(v + grp * 8) * 100 + nt * 16 + row] = acc[nt][v];
    }
    __syncthreads();
    {
        const float* tr  = shB + wave * 1600 + row * 100;   // this lane's edge row
        const float* b2p = basis2 + (size_t)edge * 8;
        v4f b20 = *(const v4f*)(b2p);          // p=0, d=0..3
        v4f b21 = *(const v4f*)(b2p + 4);      // p=1, d=0..3
        float sc0 = 0.f, sc1 = 0.f;            // heads 2*grp, 2*grp+1
        v4f vvv[8];
        #pragma unroll
        for (int oi = 0; oi < 8; ++oi) {
            int ol = grp * 8 + oi;
            v2f tk  = *(const v2f*)(tr + 2 * ol);           // conv k-chunk  (o = ol)
            v2f tq  = *(const v2f*)(tr + 2 * (16 + ol));    // conv q-chunk
            v2f tv2 = *(const v2f*)(tr + 2 * (32 + ol));    // conv v-chunk
            float dot = 0.f;
            #pragma unroll
            for (int d = 0; d < 4; ++d) {
                float p0 = b20[d], p1 = b21[d];
                float kd = tk.x * p0 + tk.y * p1;
                float qd = tq.x * p0 + tq.y * p1;
                vvv[oi][d] = tv2.x * p0 + tv2.y * p1;
                dot += kd * qd;
            }
            if (oi < 4) sc0 += dot; else sc1 += dot;
        }
        sc0 *= 0.125f; sc1 *= 0.125f;          // TEMP = HS^-0.5
        sc0 = sc0 > 0.f ? sc0 : 0.2f * sc0;    // leaky_relu
        sc1 = sc1 > 0.f ? sc1 : 0.2f * sc1;
        v2f sv; sv.x = sc0; sv.y = sc1;
        *(v2f*)(scOut + (size_t)edge * 4 + grp * 2) = sv;
        int dn = dstIdx[edge];
        atomicMax(nodeMax + dn * 4 + grp * 2,     ordf(sc0));
        atomicMax(nodeMax + dn * 4 + grp * 2 + 1, ordf(sc1));
        float* vvp = vvOut + (size_t)edge * 64 + grp * 32;
        #pragma unroll
        for (int oi = 0; oi < 8; ++oi) *(v4f*)(vvp + oi * 4) = vvv[oi];
    }
}

// ---------------------------------------------------------------------------
// Edge softmax numerator + unnormalized aggregation (normalization commutes).
// ---------------------------------------------------------------------------
__global__ __launch_bounds__(256) void k_soft(
    const float* __restrict__ scores, const float* __restrict__ vv,
    const int* __restrict__ dstIdx, const unsigned* __restrict__ nodeMax,
    float* __restrict__ nodeSum, float* __restrict__ agg)
{
    int t = blockIdx.x * 256 + threadIdx.x;    // < E*H
    int e = t >> 2, h = t & 3;
    int dn = dstIdx[e];
    float mx = unordf(nodeMax[dn * 4 + h]);
    float w  = __expf(scores[t] - mx);
    atomicAdd(nodeSum + dn * 4 + h, w);
    const float* vp = vv + (size_t)e * 64 + h * 16;
    float* ap = agg + (size_t)dn * 64 + h * 16;
    #pragma unroll
    for (int c = 0; c < 16; ++c) atomicAdd(ap + c, w * vp[c]);
}

// ---------------------------------------------------------------------------
// Per-irrep projection: row block 0 of proj_w for d==0, block 1 for d>0.
// ---------------------------------------------------------------------------
__global__ __launch_bounds__(256) void k_proj(
    const float* __restrict__ agg, const float* __restrict__ nodeSum,
    const float* __restrict__ projW, const float* __restrict__ projB,
    float* __restrict__ out)
{
    __shared__ float xsh[4][64];
    int tid = threadIdx.x;
    int nl = tid >> 6, t64 = tid & 63;
    int n = blockIdx.x * 4 + nl;
    float s = nodeSum[n * 4 + (t64 >> 4)];
    float a = agg[(size_t)n * 64 + t64];
    xsh[nl][t64] = (s > 0.f) ? (a / s) : 0.f;  // isolated nodes -> 0 (matches segment ops)
    __syncthreads();
    int m = t64 >> 2, d = t64 & 3;
    const float* wr = projW + (size_t)((d == 0 ? m : 16 + m)) * 16;
    float accv = (d == 0) ? projB[m] : 0.f;
    #pragma unroll
    for (int mp = 0; mp < 16; ++mp) accv += wr[mp] * xsh[nl][mp * 4 + d];
    out[(size_t)n * 64 + t64] = accv;
}

// ---------------------------------------------------------------------------
extern "C" void kernel_launch(void* const* d_in, const int* in_sizes, int n_in,
                              void* d_out, int out_size, void* d_ws, size_t ws_size,
                              hipStream_t stream) {
    (void)in_sizes; (void)n_in; (void)out_size; (void)ws_size;
    const float* ef   = (const float*)d_in[0];
    const float* f    = (const float*)d_in[1];
    const float* b1e  = (const float*)d_in[2];
    const float* b2e  = (const float*)d_in[3];
    const int*   srcI = (const int*)d_in[4];
    const int*   dstI = (const int*)d_in[5];
    const float* w1   = (const float*)d_in[6];
    const float* bi1  = (const float*)d_in[7];
    const float* w2   = (const float*)d_in[8];
    const float* bi2  = (const float*)d_in[9];
    const float* pw   = (const float*)d_in[10];
    const float* pb   = (const float*)d_in[11];
    float* out = (float*)d_out;

    // workspace layout (bytes): ~21 MB total
    char* ws = (char*)d_ws;
    float*    w2t     = (float*)(ws + 0);            // 2112*96*4        = 811008
    float*    vv      = (float*)(ws + 811008);       // E*64*4           = 16777216
    float*    scores  = (float*)(ws + 17588224);     // E*4*4            = 1048576
    float*    nodeSum = (float*)(ws + 18636800);     // N*4*4            = 131072
    float*    agg     = (float*)(ws + 18767872);     // N*64*4           = 2097152
    unsigned* nodeMax = (unsigned*)(ws + 20865024);  // N*4*4            = 131072

    k_prep<<<792,  256, 0, stream>>>(w2, bi2, w2t);
    k_init<<<2048, 256, 0, stream>>>(nodeSum, agg, nodeMax);
    k_conv<<<512,  256, 105472, stream>>>(ef, f, b1e, b2e, srcI, dstI, w1, bi1,
                                          w2t, vv, scores, nodeMax);
    k_soft<<<1024, 256, 0, stream>>>(scores, vv, dstI, nodeMax, nodeSum, agg);
    k_proj<<<2048, 256, 0, stream>>>(agg, nodeSum, pw, pb, out);
}